// GraphConvolution_12627203850541
// MI455X (gfx1250) — compile-verified
//
#include <hip/hip_runtime.h>
#include <hip/hip_bf16.h>

// ---------------------------------------------------------------------------
// GraphConvolution fused kernel for MI455X (gfx1250, wave32, WMMA).
//
// d       = H_e @ p.T                       [NE]
// S       = (T * d) @ T.T                   [NV,NV]   (never materialized)
// A'      = (eye ? 1 : S) * adj_v           fused elementwise
// HW      = H_v @ weight                    [NV,FOUT]
// out0    = A' @ HW + bias                  [NV,FOUT]
// out1    = H_e (passthrough)
//
// Dominant op: fp32 K=8192 contraction -> V_WMMA_F32_16X16X4_F32.
// Block tile 64x128, wave tile 32x32 (2x2 WMMA subtiles) so the LDS-fragment
// to WMMA ratio is 1:1 (one ds_load_b64 per v_wmma).
// ---------------------------------------------------------------------------

#define NVv 4096
#define NEe 8192
#define FVv 128
#define FEe 64
#define FOUT 128

typedef __attribute__((ext_vector_type(2))) float v2f;
typedef __attribute__((ext_vector_type(8))) float v8f;

static __device__ __forceinline__ v8f wmma_f32_k4(v2f a, v2f b, v8f c) {
  // 8-arg form: (neg_a, A, neg_b, B, c_mod, C, reuse_a, reuse_b)
  return __builtin_amdgcn_wmma_f32_16x16x4_f32(false, a, false, b, (short)0, c,
                                               false, false);
}

// ------------------------- small prep kernels ------------------------------

__global__ __launch_bounds__(256) void k_compute_d(const float* __restrict__ He,
                                                   const float* __restrict__ p,
                                                   float* __restrict__ dvec) {
  int k = blockIdx.x * 256 + threadIdx.x;
  if (k >= NEe) return;
  const float* row = He + (size_t)k * FEe;
  float s = 0.f;
#pragma unroll
  for (int f = 0; f < FEe; ++f) s += row[f] * p[f];
  dvec[k] = s;
}

__global__ __launch_bounds__(256) void k_compute_hw(const float* __restrict__ Hv,
                                                    const float* __restrict__ W,
                                                    float* __restrict__ HW) {
  int idx = blockIdx.x * 256 + threadIdx.x;  // NV*FOUT threads
  int i = idx >> 7;
  int j = idx & 127;
  const float* row = Hv + (size_t)i * FVv;
  float s = 0.f;
#pragma unroll 8
  for (int k = 0; k < FVv; ++k) s += row[k] * W[(size_t)k * FOUT + j];
  HW[idx] = s;
}

__global__ __launch_bounds__(256) void k_bias_init(const float* __restrict__ bias,
                                                   float* __restrict__ out) {
  int idx = blockIdx.x * 256 + threadIdx.x;  // NV*FOUT threads
  out[idx] = bias[idx & 127];
}

// ------------------------------ fused GEMM ---------------------------------

#define BM 64
#define BN 128
#define BK 32
#define LDA 36    // padded LDS stride (floats): 16B-aligned rows, conflict-free
#define LDSS 130  // padded LDS stride for masked S tile (even, 2L bank spread)
#define JT_PER_BLOCK 4

__global__ __launch_bounds__(256) void k_fused(const float* __restrict__ Tm,
                                               const float* __restrict__ dvec,
                                               const float* __restrict__ adj_v,
                                               const float* __restrict__ HW,
                                               float* __restrict__ out) {
  __shared__ float As[BM * LDA];    // (T*d) tile      [64 x 32]
  __shared__ float Bs[BN * LDA];    // T tile (rows j) [128 x 32]
  __shared__ float Ss[BM * LDSS];   // masked S' tile  [64 x 128]

  const int tid = threadIdx.x;
  const int wave = tid >> 5;
  const int lane = tid & 31;
  const int half = (lane >> 4) & 1;  // K sub-pair selector for f32 WMMA frags
  const int l16 = lane & 15;

  const int Ibase = blockIdx.x * BM;
  const int jchunk = blockIdx.y;

  // gemm1: wave -> 32x32 super-tile (2x2 subtiles of 16x16).
  const int wr = wave >> 2;   // 0..1 : rows wr*32 .. wr*32+31
  const int wc = wave & 3;    // 0..3 : cols wc*32 .. wc*32+31

  // gemm2: out tile 64x128 = 4x8 subtiles; wave -> row rm, cols cb..cb+3.
  const int rm = wave >> 1;
  const int cb = (wave & 1) * 4;

  v8f outAcc[4];
#pragma unroll
  for (int t = 0; t < 4; ++t)
    outAcc[t] = v8f{0.f, 0.f, 0.f, 0.f, 0.f, 0.f, 0.f, 0.f};

  // staging decomposition
  const int arow = tid >> 2;        // 0..63   A rows
  const int acol = (tid & 3) * 8;   // 0,8,16,24
  const int brow = tid >> 1;        // 0..127  B rows
  const int bcol = (tid & 1) * 16;  // 0,16

  for (int jt = 0; jt < JT_PER_BLOCK; ++jt) {
    const int Jbase = (jchunk * JT_PER_BLOCK + jt) * BN;

    v8f sAcc[2][2];
#pragma unroll
    for (int si = 0; si < 2; ++si)
#pragma unroll
      for (int sj = 0; sj < 2; ++sj)
        sAcc[si][sj] = v8f{0.f, 0.f, 0.f, 0.f, 0.f, 0.f, 0.f, 0.f};

    for (int k0 = 0; k0 < NEe; k0 += BK) {
      __syncthreads();  // previous compute done reading As/Bs

      // Stage A = T[Ibase..][k0..] * d (scaled at store) --------------------
      {
        const float* tsrc = Tm + (size_t)(Ibase + arow) * NEe + k0 + acol;
        const float4* ap = reinterpret_cast<const float4*>(tsrc);
        const float4* dp = reinterpret_cast<const float4*>(dvec + k0 + acol);
        float4 a0 = ap[0], a1 = ap[1];
        float4 d0 = dp[0], d1 = dp[1];
        float4 r0, r1;
        r0.x = a0.x * d0.x; r0.y = a0.y * d0.y;
        r0.z = a0.z * d0.z; r0.w = a0.w * d0.w;
        r1.x = a1.x * d1.x; r1.y = a1.y * d1.y;
        r1.z = a1.z * d1.z; r1.w = a1.w * d1.w;
        float4* ad = reinterpret_cast<float4*>(&As[arow * LDA + acol]);
        ad[0] = r0;
        ad[1] = r1;
        __builtin_prefetch(tsrc + BK, 0, 1);  // global_prefetch next K slab
      }
      // Stage B = T[Jbase..][k0..] -----------------------------------------
      {
        const float* tsrc = Tm + (size_t)(Jbase + brow) * NEe + k0 + bcol;
        const float4* bp = reinterpret_cast<const float4*>(tsrc);
        float4 b0 = bp[0], b1 = bp[1], b2 = bp[2], b3 = bp[3];
        float4* bd = reinterpret_cast<float4*>(&Bs[brow * LDA + bcol]);
        bd[0] = b0; bd[1] = b1; bd[2] = b2; bd[3] = b3;
        __builtin_prefetch(tsrc + BK, 0, 1);
      }
      __syncthreads();

      // gemm1 inner loop: 4 WMMAs per K-step (2x2), 1 ds_load_b64 per WMMA.
#pragma unroll
      for (int k = 0; k < BK; k += 4) {
        const int ak = k + 2 * half;
        v2f a0 = *reinterpret_cast<const v2f*>(
            &As[(wr * 32 + l16) * LDA + ak]);
        v2f a1 = *reinterpret_cast<const v2f*>(
            &As[(wr * 32 + 16 + l16) * LDA + ak]);
        v2f b0 = *reinterpret_cast<const v2f*>(
            &Bs[(wc * 32 + l16) * LDA + ak]);
        v2f b1 = *reinterpret_cast<const v2f*>(
            &Bs[(wc * 32 + 16 + l16) * LDA + ak]);
        sAcc[0][0] = wmma_f32_k4(a0, b0, sAcc[0][0]);
        sAcc[0][1] = wmma_f32_k4(a0, b1, sAcc[0][1]);
        sAcc[1][0] = wmma_f32_k4(a1, b0, sAcc[1][0]);
        sAcc[1][1] = wmma_f32_k4(a1, b1, sAcc[1][1]);
      }
    }

    // Eye-mask + adj_v, stage S' into LDS (C layout: M = v + 8*half, N = l16)
#pragma unroll
    for (int si = 0; si < 2; ++si) {
#pragma unroll
      for (int sj = 0; sj < 2; ++sj) {
#pragma unroll
        for (int v = 0; v < 8; ++v) {
          const int mi = wr * 32 + si * 16 + v + half * 8;
          const int nj = wc * 32 + sj * 16 + l16;
          const int gi = Ibase + mi;
          const int gj = Jbase + nj;
          float s = (gi == gj) ? 1.0f : sAcc[si][sj][v];
          Ss[mi * LDSS + nj] = s * adj_v[(size_t)gi * NVv + gj];
        }
      }
    }
    __syncthreads();

    // gemm2: outAcc += S'(64x128) @ HW[Jbase:Jbase+128, 0:128]
#pragma unroll 4
    for (int k = 0; k < BN; k += 4) {
      const int ak = k + 2 * half;
      v2f a = *reinterpret_cast<const v2f*>(&Ss[(rm * 16 + l16) * LDSS + ak]);
#pragma unroll
      for (int t = 0; t < 4; ++t) {
        const int col = (cb + t) * 16 + l16;
        v2f b;
        b.x = HW[(size_t)(Jbase + ak) * FOUT + col];
        b.y = HW[(size_t)(Jbase + ak + 1) * FOUT + col];
        outAcc[t] = wmma_f32_k4(a, b, outAcc[t]);
      }
    }
  }

  // Reduce partial output slice into out (bias pre-initialized).
#pragma unroll
  for (int t = 0; t < 4; ++t) {
    const int col = (cb + t) * 16 + l16;
#pragma unroll
    for (int v = 0; v < 8; ++v) {
      const int row = Ibase + rm * 16 + v + half * 8;
      atomicAdd(&out[(size_t)row * FOUT + col], outAcc[t][v]);
    }
  }
}

// ------------------------------- launcher ----------------------------------

extern "C" void kernel_launch(void* const* d_in, const int* in_sizes, int n_in,
                              void* d_out, int out_size, void* d_ws,
                              size_t ws_size, hipStream_t stream) {
  (void)in_sizes; (void)n_in; (void)out_size; (void)ws_size;

  const float* H_v   = (const float*)d_in[0];
  const float* H_e   = (const float*)d_in[1];
  // d_in[2] = adj_e (unused by node_layer path)
  const float* adj_v = (const float*)d_in[3];
  const float* Tm    = (const float*)d_in[4];
  const float* W     = (const float*)d_in[5];
  const float* p     = (const float*)d_in[6];
  const float* bias  = (const float*)d_in[7];

  float* out = (float*)d_out;
  float* wsf = (float*)d_ws;
  float* dvec = wsf;            // NE floats
  float* HW   = wsf + NEe;      // NV*FOUT floats

  k_compute_d<<<NEe / 256, 256, 0, stream>>>(H_e, p, dvec);
  k_compute_hw<<<(NVv * FOUT) / 256, 256, 0, stream>>>(H_v, W, HW);
  k_bias_init<<<(NVv * FOUT) / 256, 256, 0, stream>>>(bias, out);

  // Second tuple output: H_e passthrough.
  hipMemcpyAsync(out + (size_t)NVv * FOUT, H_e, sizeof(float) * NEe * FEe,
                 hipMemcpyDeviceToDevice, stream);

  dim3 grid(NVv / BM, (NVv / BN) / JT_PER_BLOCK);  // 64 x 8 = 512 workgroups
  k_fused<<<grid, 256, 0, stream>>>(Tm, dvec, adj_v, HW, out);
}